// Partition_Linear_row_69569880261337
// MI455X (gfx1250) — compile-verified
//
#include <hip/hip_runtime.h>
#include <stdint.h>

// Problem constants (from reference): input (8192, 4096) f32, weight (4096, 4096) f32.
#define IN_F   4096
#define OUT_F  4096
#define BATCH  8192

typedef float v4f __attribute__((ext_vector_type(4)));
typedef int   v4i __attribute__((ext_vector_type(4)));

// Typed address-space pointers for the gfx1250 async global->LDS builtin.
// HIP diagnostics print as(1) as `__device__`: param0 is `v4i addrspace(1)*`
// (non-const), param1 is the LDS-side pointer.
typedef __attribute__((address_space(1))) v4i gv4i;
typedef __attribute__((address_space(3))) v4i lv4i;

// ---- Detect gfx1250 async global->LDS builtins (guarded: file compiles either way) ----
#if defined(__HIP_DEVICE_COMPILE__) && __has_builtin(__builtin_amdgcn_global_load_async_to_lds_b128)
#define HAVE_ASYNC_LDS 1
#else
#define HAVE_ASYNC_LDS 0
#endif

__device__ __forceinline__ uint32_t f2ord(float f) {
    // Monotonic float->uint mapping: a > b (floats) <=> f2ord(a) > f2ord(b) (uints).
    uint32_t u = __float_as_uint(f);
    return (u & 0x80000000u) ? ~u : (u | 0x80000000u);
}

// ---------------------------------------------------------------------------
// Kernel 0: reset packed argmax keys (must run every call: keys accumulate via
// atomics and the harness does not re-poison between graph replays).
// ---------------------------------------------------------------------------
__global__ void k_init_keys(unsigned long long* __restrict__ keys) {
    int j = blockIdx.x * blockDim.x + threadIdx.x;
    if (j < IN_F) keys[j] = 0ull;   // 0 < f2ord(x)<<32 for any finite x in range
}

// ---------------------------------------------------------------------------
// Kernel 1: column argmax of weight (OUT_F rows x IN_F cols, row-major).
// grid = (IN_F/1024, 64), block = 256. Each thread: 4 consecutive columns,
// 64 rows, b128 nontemporal loads (weight is streamed exactly once).
// Publishes per-chunk winner with one u64 atomic-max per column.
// Key = (f2ord(val) << 32) | ~row  -> max picks largest val, then SMALLEST row
// on exact ties (matches jnp.argmax first-occurrence).
// ---------------------------------------------------------------------------
#define A_ROWS (OUT_F / 64)   // 64 rows per chunk

__global__ __launch_bounds__(256) void k_argmax(const float* __restrict__ W,
                                                unsigned long long* __restrict__ keys) {
    const int tid = threadIdx.x;
    const int j0  = blockIdx.x * 1024 + tid * 4;
    const int r0  = blockIdx.y * A_ROWS;

    float b0 = -__builtin_inff(), b1 = b0, b2 = b0, b3 = b0;
    int   i0 = 0, i1 = 0, i2 = 0, i3 = 0;

    const char* base = (const char*)W + ((size_t)r0 * IN_F + j0) * sizeof(float);
#pragma unroll 4
    for (int r = 0; r < A_ROWS; ++r) {
        v4f v = __builtin_nontemporal_load((const v4f*)(base + (size_t)r * (IN_F * sizeof(float))));
        const int row = r0 + r;
        if (v.x > b0) { b0 = v.x; i0 = row; }
        if (v.y > b1) { b1 = v.y; i1 = row; }
        if (v.z > b2) { b2 = v.z; i2 = row; }
        if (v.w > b3) { b3 = v.w; i3 = row; }
    }

    atomicMax(&keys[j0 + 0], ((unsigned long long)f2ord(b0) << 32) | (uint32_t)~(uint32_t)i0);
    atomicMax(&keys[j0 + 1], ((unsigned long long)f2ord(b1) << 32) | (uint32_t)~(uint32_t)i1);
    atomicMax(&keys[j0 + 2], ((unsigned long long)f2ord(b2) << 32) | (uint32_t)~(uint32_t)i2);
    atomicMax(&keys[j0 + 3], ((unsigned long long)f2ord(b3) << 32) | (uint32_t)~(uint32_t)i3);
}

// ---------------------------------------------------------------------------
// Kernel 2: unpack row index from packed key.
// ---------------------------------------------------------------------------
__global__ void k_decode(const unsigned long long* __restrict__ keys, int* __restrict__ ii) {
    int j = blockIdx.x * blockDim.x + threadIdx.x;
    if (j < IN_F) ii[j] = (int)(~(uint32_t)keys[j]);
}

// ---------------------------------------------------------------------------
// Kernel 3: gather  y[b, j] = x[b, ii[j]].
// grid = (IN_F/1024, BATCH/8), block = 256. Thread -> 4 consecutive columns,
// block -> 8 rows (indices reused 8x). Index slice staged into LDS with the
// gfx1250 async global->LDS path (ASYNCcnt-tracked DMA). Scattered b32
// gathers hit a 16 KB input-row region that stays resident in WGP$/L2;
// output stores are nontemporal b128 (write-once stream, spare L2 for reuse).
// ---------------------------------------------------------------------------
#define G_ROWS 8

__global__ __launch_bounds__(256) void k_gather(const float* __restrict__ x,
                                                const int* __restrict__ ii,
                                                float* __restrict__ y) {
    __shared__ int lds_idx[256 * 4];   // 4 KB index stage
    const int tid = threadIdx.x;
    const int j0  = blockIdx.x * 1024 + tid * 4;

#if HAVE_ASYNC_LDS
    // Async DMA: global (ii + j0, 16B/lane) -> LDS, tracked by ASYNCcnt.
    // C-style casts perform the addrspacecasts (flat->global is a no-op value
    // change; flat->LDS strips the aperture base).
    __builtin_amdgcn_global_load_async_to_lds_b128(
        (gv4i*)(ii + j0),
        (lv4i*)(&lds_idx[tid * 4]),
        /*offset=*/0, /*cpol=*/0);
#if __has_builtin(__builtin_amdgcn_s_wait_asynccnt)
    __builtin_amdgcn_s_wait_asynccnt(0);
#else
    asm volatile("s_wait_asynccnt 0" ::: "memory");
#endif
    __syncthreads();
#else
    *(v4i*)(&lds_idx[tid * 4]) = *(const v4i*)(ii + j0);
    __syncthreads();
#endif

    const int i0 = lds_idx[tid * 4 + 0];
    const int i1 = lds_idx[tid * 4 + 1];
    const int i2 = lds_idx[tid * 4 + 2];
    const int i3 = lds_idx[tid * 4 + 3];

    const size_t b0 = (size_t)blockIdx.y * G_ROWS;
    const float* xr = x + b0 * IN_F;
    float*       yr = y + b0 * IN_F + j0;

#pragma unroll
    for (int r = 0; r < G_ROWS; ++r) {
        const float* row = xr + (size_t)r * IN_F;
        v4f v;
        v.x = row[i0];
        v.y = row[i1];
        v.z = row[i2];
        v.w = row[i3];
        __builtin_nontemporal_store(v, (v4f*)(yr + (size_t)r * IN_F));
    }
}

// ---------------------------------------------------------------------------
// Host launcher. d_in[0] = input (BATCH*OUT_F f32), d_in[1] = weight
// (OUT_F*IN_F f32), d_out = BATCH*IN_F f32. Workspace: 32 KB keys + 16 KB ii.
// ---------------------------------------------------------------------------
extern "C" void kernel_launch(void* const* d_in, const int* in_sizes, int n_in,
                              void* d_out, int out_size, void* d_ws, size_t ws_size,
                              hipStream_t stream) {
    const float* x = (const float*)d_in[0];
    const float* W = (const float*)d_in[1];
    float*       y = (float*)d_out;

    unsigned long long* keys = (unsigned long long*)d_ws;
    int* ii = (int*)((char*)d_ws + (size_t)IN_F * sizeof(unsigned long long));

    k_init_keys<<<IN_F / 256, 256, 0, stream>>>(keys);

    dim3 ga(IN_F / 1024, OUT_F / A_ROWS);   // (4, 64)
    k_argmax<<<ga, 256, 0, stream>>>(W, keys);

    k_decode<<<IN_F / 256, 256, 0, stream>>>(keys, ii);

    dim3 gg(IN_F / 1024, BATCH / G_ROWS);   // (4, 1024)
    k_gather<<<gg, 256, 0, stream>>>(x, ii, y);
}